// Locally_connected2D_53051436040756
// MI455X (gfx1250) — compile-verified
//
#include <hip/hip_runtime.h>

typedef __attribute__((ext_vector_type(2))) float v2f;
typedef __attribute__((ext_vector_type(8))) float v8f;

#define KDIM   75
#define KPAD   76
#define BATCH  128
#define OCH    32
#define NPOS   3600   // 60*60

#if defined(__has_builtin)
#if __has_builtin(__builtin_amdgcn_global_load_async_to_lds_b32) && \
    __has_builtin(__builtin_amdgcn_s_wait_asynccnt)
#define USE_ASYNC_LDS 1
#endif
#endif

#ifdef USE_ASYNC_LDS
typedef __attribute__((address_space(1))) int as1_int;   // global (AS1)
typedef __attribute__((address_space(3))) int as3_int;   // LDS    (AS3)
#endif

__global__ __launch_bounds__(256) void lc2d_wmma_f32(
    const float* __restrict__ x,     // [128,3,64,64]
    const float* __restrict__ W,     // [60,60,32,75]
    const float* __restrict__ bias,  // [60,60,32]
    float* __restrict__ out)         // [128,32,60,60]
{
    __shared__ float P[BATCH * KPAD];  // patches  [b][k], k padded with 0
    __shared__ float Wl[OCH * KPAD];   // weights  [o][k], k padded with 0

    const int pos = blockIdx.x;        // 0..3599
    const int py  = pos / 60;
    const int px  = pos - py * 60;
    const int tid = threadIdx.x;

    const float* Wp = W + (size_t)pos * (OCH * KDIM);

#ifdef USE_ASYNC_LDS
    // ---- zero-fill the k==75 pad column (async loads can't synthesize it) ----
    if (tid < BATCH) P[tid * KPAD + KDIM]  = 0.0f;
    if (tid < OCH)   Wl[tid * KPAD + KDIM] = 0.0f;

    // ---- async stage weight slice (contiguous [32][75]) into LDS ----
    for (int e = tid; e < OCH * KDIM; e += 256) {
        int o = e / KDIM;
        int k = e - o * KDIM;
        __builtin_amdgcn_global_load_async_to_lds_b32(
            (as1_int*)(Wp + e), (as3_int*)&Wl[o * KPAD + k], 0, 0);
    }

    // ---- async stage patches: P[b][k] = x[b][c][py+r][px+s], k=c*25+r*5+s ----
    for (int e = tid; e < BATCH * KDIM; e += 256) {
        int b  = e / KDIM;
        int k  = e - b * KDIM;
        int c  = k / 25;
        int rs = k - c * 25;
        int r  = rs / 5;
        int s  = rs - r * 5;
        const float* gp = &x[((b * 3 + c) * 64 + (py + r)) * 64 + (px + s)];
        __builtin_amdgcn_global_load_async_to_lds_b32(
            (as1_int*)gp, (as3_int*)&P[b * KPAD + k], 0, 0);
    }

    __builtin_amdgcn_s_wait_asynccnt(0);
    __syncthreads();
#else
    // ---- fallback: synchronous staging through VGPRs ----
    for (int e = tid; e < OCH * KPAD; e += 256) {
        int o = e / KPAD;
        int k = e - o * KPAD;
        Wl[e] = (k < KDIM) ? Wp[o * KDIM + k] : 0.0f;
    }
    for (int e = tid; e < BATCH * KPAD; e += 256) {
        int b = e / KPAD;
        int k = e - b * KPAD;
        float v = 0.0f;
        if (k < KDIM) {
            int c  = k / 25;
            int rs = k - c * 25;
            int r  = rs / 5;
            int s  = rs - r * 5;
            v = x[((b * 3 + c) * 64 + (py + r)) * 64 + (px + s)];
        }
        P[e] = v;
    }
    __syncthreads();
#endif

    // ---- per-wave 16x32 output tile via V_WMMA_F32_16X16X4_F32 ----
    const int wave = tid >> 5;        // 0..7 -> M tile
    const int lane = tid & 31;
    const int llo  = lane & 15;
    const int half = lane >> 4;       // K split for A/B frags

    const int mb = wave * 16;

    v8f acc0 = {0.f, 0.f, 0.f, 0.f, 0.f, 0.f, 0.f, 0.f};
    v8f acc1 = {0.f, 0.f, 0.f, 0.f, 0.f, 0.f, 0.f, 0.f};

    // A frag: lane<16 -> (M=llo, K=k0..k0+1); lane>=16 -> (M=llo, K=k0+2..k0+3)
    const float* Pa  = &P[(mb + llo) * KPAD + half * 2];
    // B frag: lane<16 -> (N=llo, K=k0..k0+1); lane>=16 -> (N=llo, K=k0+2..k0+3)
    const float* Bb0 = &Wl[llo * KPAD + half * 2];
    const float* Bb1 = &Wl[(llo + 16) * KPAD + half * 2];

#pragma unroll
    for (int kk = 0; kk < 19; ++kk) {
        const int k0 = kk * 4;
        v2f a  = *(const v2f*)(Pa  + k0);
        v2f b0 = *(const v2f*)(Bb0 + k0);
        v2f b1 = *(const v2f*)(Bb1 + k0);
        acc0 = __builtin_amdgcn_wmma_f32_16x16x4_f32(
            false, a, false, b0, (short)0, acc0, false, false);
        acc1 = __builtin_amdgcn_wmma_f32_16x16x4_f32(
            false, a, false, b1, (short)0, acc1, false, false);
    }

    // ---- bias + store. C/D layout: lane -> N=llo, rows M = half*8 + j ----
    const float bia0 = bias[pos * OCH + llo];
    const float bia1 = bias[pos * OCH + 16 + llo];

    const int m0 = mb + half * 8;
#pragma unroll
    for (int j = 0; j < 8; ++j) {
        const int m = m0 + j;
        out[(size_t)(m * OCH + llo)      * NPOS + pos] = acc0[j] + bia0;
        out[(size_t)(m * OCH + 16 + llo) * NPOS + pos] = acc1[j] + bia1;
    }
}

extern "C" void kernel_launch(void* const* d_in, const int* in_sizes, int n_in,
                              void* d_out, int out_size, void* d_ws, size_t ws_size,
                              hipStream_t stream) {
    const float* x = (const float*)d_in[0];   // 128*3*64*64
    const float* W = (const float*)d_in[1];   // 60*60*32*75
    const float* b = (const float*)d_in[2];   // 60*60*32
    float* out = (float*)d_out;               // 128*32*60*60

    lc2d_wmma_f32<<<dim3(NPOS), dim3(256), 0, stream>>>(x, W, b, out);
}